// JaxLSHRouter_29154238005386
// MI455X (gfx1250) — compile-verified
//
#include <hip/hip_runtime.h>

typedef __attribute__((ext_vector_type(2))) float v2f;
typedef __attribute__((ext_vector_type(8))) float v8f;

#define NUM_EXPERTS 8
#define TOP_K 2
#define D_DIM 2048
#define N_TOK 4096
#define CAP 512
#define NEC (N_TOK * NUM_EXPERTS * CAP) /* 16777216 dispatcher elements */
#define NPAD 16                          /* experts padded to WMMA N=16 */

// ---------------------------------------------------------------------------
// Kernel P: prepack W [2048,8] -> Wp [1024 pairs][16 experts][2 K-values],
// zero-padded to 16 experts. Lane-contiguous b64 B-fragment loads in the GEMM.
// ---------------------------------------------------------------------------
__global__ __launch_bounds__(256) void k_prepack(const float* __restrict__ W,
                                                 float2* __restrict__ Wp) {
    const int idx = blockIdx.x * blockDim.x + threadIdx.x; // 0..16383
    const int p = idx >> 4;   // K-pair index (kk = 2p)
    const int n = idx & 15;   // padded expert column
    float2 v = make_float2(0.f, 0.f);
    if (n < NUM_EXPERTS) {
        v.x = W[(size_t)(2 * p + 0) * NUM_EXPERTS + n];
        v.y = W[(size_t)(2 * p + 1) * NUM_EXPERTS + n];
    }
    Wp[p * NPAD + n] = v;
}

// ---------------------------------------------------------------------------
// Kernel Z: zero the dispatcher region (64 MB) with wide stores.
// ---------------------------------------------------------------------------
__global__ __launch_bounds__(256) void k_zero(float4* __restrict__ p, int n4) {
    int i = blockIdx.x * blockDim.x + threadIdx.x;
    int stride = gridDim.x * blockDim.x;
    float4 z = make_float4(0.f, 0.f, 0.f, 0.f);
    for (; i < n4; i += stride) p[i] = z;
}

// ---------------------------------------------------------------------------
// Kernel 1: logits GEMM via V_WMMA_F32_16X16X4_F32 + top-2 gating + softmax.
// 2 waves per 16-token tile, K split in half across waves (512 waves total).
//
// A (16x4 f32): lane l: M = l&15, khalf = l>>4; VGPR v holds K = k0+2*khalf+v.
// B (4x16 f32): lane l: N = l&15, khalf = l>>4; VGPR v holds K = k0+2*khalf+v.
// C/D (16x16 f32): lane l: N = l&15, mhalf = l>>4; VGPR r holds M = r+8*mhalf.
// ---------------------------------------------------------------------------
__global__ __launch_bounds__(64) void k_gemm_gate(const float* __restrict__ x,
                                                  const float* __restrict__ Wp,
                                                  float* __restrict__ out,
                                                  int* __restrict__ eidx_ws) {
    __shared__ float lsc[2][16][NUM_EXPERTS]; // per-wave partial logits
    const int lane  = threadIdx.x & 31;
    const int wv    = threadIdx.x >> 5;       // K-half owner
    const int row0  = blockIdx.x * 16;

    const int m     = lane & 15;
    const int khalf = lane >> 4;
    const int n     = lane & 15;

    const int kbeg = wv * (D_DIM / 2);
    const float* xrow = x + (size_t)(row0 + m) * D_DIM + kbeg + 2 * khalf;
    // B: pair index p = (kbeg+k0)/2 + khalf, element offset p*32 + n*2 floats
    const float* wp = Wp + ((size_t)(kbeg >> 1) + khalf) * (2 * NPAD) + 2 * n;

    v8f c = {};
    #pragma unroll 8
    for (int k0 = 0; k0 < D_DIM / 2; k0 += 4) {
        v2f a;
        a.x = xrow[k0];
        a.y = xrow[k0 + 1];
        const float* wq = wp + (k0 >> 1) * (2 * NPAD);
        v2f b;
        b.x = wq[0];
        b.y = wq[1];
        // (neg_a, A, neg_b, B, c_mod, C, reuse_a, reuse_b)
        c = __builtin_amdgcn_wmma_f32_16x16x4_f32(false, a, false, b,
                                                  (short)0, c, false, false);
    }

    // C-fragment (lane = expert) -> token-major partial logits in LDS.
    if (n < NUM_EXPERTS) {
        #pragma unroll
        for (int r = 0; r < 8; ++r)
            lsc[wv][r + 8 * khalf][n] = c[r];
    }
    __syncthreads();

    if (threadIdx.x < 16) {
        const int t   = threadIdx.x;
        const int tok = row0 + t;
        float l[NUM_EXPERTS];
        #pragma unroll
        for (int e = 0; e < NUM_EXPERTS; ++e)
            l[e] = lsc[0][t][e] + lsc[1][t][e];

        // top-2, ties -> lower index (strict > so first occurrence wins)
        int b0 = 0; float v0 = l[0];
        #pragma unroll
        for (int e = 1; e < NUM_EXPERTS; ++e)
            if (l[e] > v0) { v0 = l[e]; b0 = e; }
        int b1 = (b0 == 0) ? 1 : 0; float v1 = l[b1];
        #pragma unroll
        for (int e = 0; e < NUM_EXPERTS; ++e)
            if (e != b0 && l[e] > v1) { v1 = l[e]; b1 = e; }

        // softmax over the two (descending) gate logits
        const float ex  = __expf(v1 - v0);
        const float inv = 1.0f / (1.0f + ex);
        float* gate = out + NEC;
        float* eif  = out + NEC + N_TOK * TOP_K;
        gate[tok * 2 + 0] = inv;
        gate[tok * 2 + 1] = ex * inv;
        eif[tok * 2 + 0]  = (float)b0;
        eif[tok * 2 + 1]  = (float)b1;
        eidx_ws[tok * 2 + 0] = b0;
        eidx_ws[tok * 2 + 1] = b1;
    }
}

// ---------------------------------------------------------------------------
// Kernel 2: fused scan + scatter. One 256-thread block.
// Phase A: stage all expert indices into LDS (coalesced).
// Phase B: wave 0 runs the order-exact ballot scan (cumsum per (k,e) channel)
//          against LDS, overwriting the staged entries with packed slot codes
//          (slot<<4)|(expert<<1)|valid.
// Phase C: all 8 waves scatter 1.0f into the pre-zeroed dispatcher.
// ---------------------------------------------------------------------------
__global__ __launch_bounds__(256) void k_scan_scatter(const int* __restrict__ eidx,
                                                      float* __restrict__ disp) {
    __shared__ int se[N_TOK * TOP_K]; // 32 KB: indices, then slot codes
    const int tid = threadIdx.x;

    for (int i = tid; i < N_TOK * TOP_K; i += 256)
        se[i] = eidx[i];
    __syncthreads();

    if (tid < 32) {
        const int lane = tid;
        const unsigned lmask = (1u << lane) - 1u; // lanes strictly below
        int cnt[TOP_K][NUM_EXPERTS];
        #pragma unroll
        for (int k = 0; k < TOP_K; ++k)
            #pragma unroll
            for (int e = 0; e < NUM_EXPERTS; ++e) cnt[k][e] = 0;

        for (int i = 0; i < N_TOK / 32; ++i) {
            const int tok = i * 32 + lane;
            int ek[TOP_K];
            ek[0] = se[tok * 2 + 0];
            ek[1] = se[tok * 2 + 1];
            #pragma unroll
            for (int k = 0; k < TOP_K; ++k) {
                int myslot = 0, myvalid = 0;
                #pragma unroll
                for (int e = 0; e < NUM_EXPERTS; ++e) {
                    const unsigned m32 = (unsigned)__ballot(ek[k] == e);
                    if (ek[k] == e) {
                        const int rank = __popc(m32 & lmask);
                        const int prio = cnt[k][e] + rank + 1; // inclusive cumsum
                        myvalid = (prio <= CAP) ? 1 : 0;
                        myslot  = prio - 1;
                    }
                    cnt[k][e] += __popc(m32);
                }
                se[tok * 2 + k] = (myslot << 4) | (ek[k] << 1) | myvalid;
            }
        }
    }
    __syncthreads();

    for (int i = tid; i < N_TOK * TOP_K; i += 256) {
        const int v = se[i];
        if (v & 1) {
            const int tok  = i >> 1;
            const int e    = (v >> 1) & 7;
            const int slot = v >> 4;
            disp[(size_t)tok * NUM_EXPERTS * CAP + e * CAP + slot] = 1.0f;
        }
    }
}

extern "C" void kernel_launch(void* const* d_in, const int* in_sizes, int n_in,
                              void* d_out, int out_size, void* d_ws, size_t ws_size,
                              hipStream_t stream) {
    const float* x = (const float*)d_in[0]; // [2,2048,2048] f32
    const float* W = (const float*)d_in[1]; // [2048,8] f32
    float* out = (float*)d_out;

    float* Wp  = (float*)d_ws;                              // 2048*16 f32 = 128 KB
    int*   eidx = (int*)((char*)d_ws + (size_t)D_DIM * NPAD * sizeof(float)); // 32 KB

    k_prepack<<<(D_DIM / 2 * NPAD) / 256, 256, 0, stream>>>(W, (float2*)Wp);
    k_gemm_gate<<<N_TOK / 16, 64, 0, stream>>>(x, Wp, out, eidx);
    k_zero<<<4096, 256, 0, stream>>>((float4*)out, NEC / 4);
    k_scan_scatter<<<1, 256, 0, stream>>>(eidx, out);
}